// MyModel_44023414784638
// MI455X (gfx1250) — compile-verified
//
#include <hip/hip_runtime.h>

#define D_MODEL    1024
#define SEQ_LEN    32768
#define CHUNK_ROWS 512
#define NUM_CHUNKS (SEQ_LEN / CHUNK_ROWS)   /* 64 */
#define TWO_PI     6.28318530717958647692f

typedef float v2f __attribute__((ext_vector_type(2)));
typedef float v8f __attribute__((ext_vector_type(8)));

// ---------------------------------------------------------------------------
// Stage 1: partial sin-reductions over L-chunks.
// grid = (NUM_CHUNKS, D_MODEL/256), block = 256 threads.
// partials[chunk][d] = sum_{l in chunk} sin(2*pi*x[l]*(t[d]+log1p(l)))
// ---------------------------------------------------------------------------
__global__ void sig_partial_kernel(const float* __restrict__ in,
                                   float* __restrict__ partials) {
    __shared__ float s_x[CHUNK_ROWS];
    __shared__ float s_p[CHUNK_ROWS];

    const int chunk = blockIdx.x;
    const int d     = blockIdx.y * blockDim.x + threadIdx.x;   // 0..1023
    const int l0    = chunk * CHUNK_ROWS;

    for (int j = threadIdx.x; j < CHUNK_ROWS; j += blockDim.x) {
        s_x[j] = in[l0 + j];
        s_p[j] = log1pf((float)(l0 + j));
    }
    __syncthreads();

    const float t = (float)d * (1.0f / (float)(D_MODEL - 1));
    float acc = 0.0f;
#pragma unroll 8
    for (int j = 0; j < CHUNK_ROWS; ++j) {
        acc += __sinf(TWO_PI * s_x[j] * (t + s_p[j]));
    }
    partials[chunk * D_MODEL + d] = acc;
}

// ---------------------------------------------------------------------------
// Stage 2: deterministic reduction of the 64 partials per column.
// ---------------------------------------------------------------------------
__global__ void sig_reduce_kernel(const float* __restrict__ partials,
                                  float* __restrict__ signal) {
    const int d = blockIdx.x * blockDim.x + threadIdx.x;
    float acc = 0.0f;
#pragma unroll 8
    for (int c = 0; c < NUM_CHUNKS; ++c) {
        acc += partials[c * D_MODEL + d];
    }
    signal[d] = acc;
}

// ---------------------------------------------------------------------------
// Stage 3: out[i] = sum_k signal[k] * W[i][k] + b[i] via V_WMMA_F32_16X16X4_F32.
// One wave32 per 16-output tile; 64 waves total (8 blocks x 256 threads).
//
// A (16x4 f32, 2 VGPRs): row m = signal[k0..k0+3] broadcast.
//   lanes 0-15:  {a.x, a.y} = {K=0, K=1};  lanes 16-31: {K=2, K=3}.
// B (4x16 f32, 2 VGPRs): B[k][n] = W[base+n][k0+k]
//   lanes 0-15 hold n=lane with {b.x,b.y} = {K=0,K=1}; lanes 16-31: {K=2,K=3}.
// D row M=0 lives in C-VGPR0 of lanes 0-15.
//
// Four independent accumulators (K interleaved by 4) so consecutive v_wmma
// have no D->C RAW hazard; loads for all four batch into clauses ahead of
// the WMMA group. Accumulators combined at the end (element 0 only).
// ---------------------------------------------------------------------------
__global__ void gemv_wmma_kernel(const float* __restrict__ signal,
                                 const float* __restrict__ W,
                                 const float* __restrict__ bias,
                                 float* __restrict__ out) {
    const int wavesPerBlock = blockDim.x >> 5;
    const int wave = blockIdx.x * wavesPerBlock + (threadIdx.x >> 5);
    const int lane = threadIdx.x & 31;
    const int half = lane >> 4;     // 0: K=0,1   1: K=2,3
    const int r    = lane & 15;     // output column within tile
    const int base = wave * 16;     // first output index of this tile

    const float* __restrict__ wrow = W + (size_t)(base + r) * D_MODEL + half * 2;
    const float* __restrict__ sp   = signal + half * 2;

    v8f c0 = {}, c1 = {}, c2 = {}, c3 = {};
#pragma unroll 2
    for (int k0 = 0; k0 < D_MODEL; k0 += 16) {
        v2f a0 = *(const v2f*)(sp + k0);
        v2f a1 = *(const v2f*)(sp + k0 + 4);
        v2f a2 = *(const v2f*)(sp + k0 + 8);
        v2f a3 = *(const v2f*)(sp + k0 + 12);
        v2f b0 = *(const v2f*)(wrow + k0);
        v2f b1 = *(const v2f*)(wrow + k0 + 4);
        v2f b2 = *(const v2f*)(wrow + k0 + 8);
        v2f b3 = *(const v2f*)(wrow + k0 + 12);
        // (neg_a, A, neg_b, B, c_mod, C, reuse_a, reuse_b)
        c0 = __builtin_amdgcn_wmma_f32_16x16x4_f32(false, a0, false, b0,
                                                   (short)0, c0, false, false);
        c1 = __builtin_amdgcn_wmma_f32_16x16x4_f32(false, a1, false, b1,
                                                   (short)0, c1, false, false);
        c2 = __builtin_amdgcn_wmma_f32_16x16x4_f32(false, a2, false, b2,
                                                   (short)0, c2, false, false);
        c3 = __builtin_amdgcn_wmma_f32_16x16x4_f32(false, a3, false, b3,
                                                   (short)0, c3, false, false);
    }

    if (lane < 16) {
        out[base + r] = (c0[0] + c1[0]) + (c2[0] + c3[0]) + bias[base + r];
    }
}

// ---------------------------------------------------------------------------
extern "C" void kernel_launch(void* const* d_in, const int* in_sizes, int n_in,
                              void* d_out, int out_size, void* d_ws, size_t ws_size,
                              hipStream_t stream) {
    (void)in_sizes; (void)n_in; (void)out_size; (void)ws_size;

    const float* d_inputs = (const float*)d_in[0];   // [32768]
    const float* d_W      = (const float*)d_in[1];   // [1024,1024]
    const float* d_b      = (const float*)d_in[2];   // [1024]
    float*       out      = (float*)d_out;           // [1024]

    float* partials = (float*)d_ws;                         // 64*1024 f32
    float* signal   = partials + NUM_CHUNKS * D_MODEL;      // 1024 f32

    dim3 g1(NUM_CHUNKS, D_MODEL / 256);
    sig_partial_kernel<<<g1, 256, 0, stream>>>(d_inputs, partials);

    sig_reduce_kernel<<<D_MODEL / 256, 256, 0, stream>>>(partials, signal);

    // 64 tiles of 16 outputs, 8 waves per 256-thread block -> 8 blocks
    gemv_wmma_kernel<<<(D_MODEL / 16) / 8, 256, 0, stream>>>(signal, d_W, d_b, out);
}